// GCN_59742995087830
// MI455X (gfx1250) — compile-verified
//
#include <hip/hip_runtime.h>
#include <math.h>

#define DD 128  // embedding dim (fixed by reference)

typedef __attribute__((ext_vector_type(2))) float v2f;
typedef __attribute__((ext_vector_type(8))) float v8f;

__device__ __forceinline__ void atomAddF32(float* p, float v) {
  // relaxed, device-scope fp32 atomic -> global_atomic_add_f32 on gfx1250
  __hip_atomic_fetch_add(p, v, __ATOMIC_RELAXED, __HIP_MEMORY_SCOPE_AGENT);
}

// ---------------- utility kernels ----------------

__global__ void zero4_kernel(float4* __restrict__ p, long n4) {
  long i = (long)blockIdx.x * blockDim.x + threadIdx.x;
  if (i < n4) p[i] = make_float4(0.f, 0.f, 0.f, 0.f);
}

__global__ void deg_kernel(const int* __restrict__ dst, float* __restrict__ deg, int E) {
  int e = blockIdx.x * blockDim.x + threadIdx.x;
  if (e < E) {
    __builtin_prefetch(dst + e + 8192, 0, 1);
    atomAddF32(&deg[dst[e]], 1.0f);
  }
}

__global__ void dinv_kernel(float* __restrict__ dinv, int N) {
  int i = blockIdx.x * blockDim.x + threadIdx.x;
  if (i < N) {
    float d = dinv[i];
    dinv[i] = (d > 0.f) ? (1.0f / sqrtf(d)) : 0.f;
  }
}

// acc = w*emb ; x = emb
__global__ void init_kernel(const float4* __restrict__ emb, float4* __restrict__ x,
                            float4* __restrict__ acc, long n4, float w) {
  long i = (long)blockIdx.x * blockDim.x + threadIdx.x;
  if (i < n4) {
    float4 v = emb[i];
    x[i] = v;
    acc[i] = make_float4(w * v.x, w * v.y, w * v.z, w * v.w);
  }
}

// acc += w*x
__global__ void axpy_kernel(float4* __restrict__ acc, const float4* __restrict__ x,
                            long n4, float w) {
  long i = (long)blockIdx.x * blockDim.x + threadIdx.x;
  if (i < n4) {
    float4 a = acc[i];
    float4 v = x[i];
    a.x += w * v.x; a.y += w * v.y; a.z += w * v.z; a.w += w * v.w;
    acc[i] = a;
  }
}

// ---------------- LGConv scatter: one wave per edge, float4 per lane ----------------

__global__ void scatter_kernel(const int* __restrict__ src, const int* __restrict__ dst,
                               const float* __restrict__ dinv,
                               const float* __restrict__ xin, float* __restrict__ xout,
                               int E) {
  long tid = (long)blockIdx.x * blockDim.x + threadIdx.x;
  int e    = (int)(tid >> 5);
  int lane = (int)(tid & 31);
  if (e >= E) return;
  __builtin_prefetch(src + e + 4096, 0, 1);
  __builtin_prefetch(dst + e + 4096, 0, 1);
  int s = src[e];
  int d = dst[e];
  float nrm = dinv[s] * dinv[d];
  const float4 v = *(const float4*)(xin + (long)s * DD + lane * 4);
  float* o = xout + (long)d * DD + lane * 4;
  atomAddF32(o + 0, nrm * v.x);
  atomAddF32(o + 1, nrm * v.y);
  atomAddF32(o + 2, nrm * v.z);
  atomAddF32(o + 3, nrm * v.w);
}

// ---------------- batched dot products via V_WMMA_F32_16X16X4_F32 ----------------
// One wave computes 16 pair-dots: C = A(16x128) * B(128x16) accumulated over 32
// K=4 WMMA steps; result pairs are the diagonal of the 16x16 f32 accumulator.
// A lane layout (16x4 f32): lane L holds row M=L%16, K = 2*(L/16)+{0,1}  -> float2 load
// B lane layout (4x16 f32): lane L holds col N=L%16, K = 2*(L/16)+{0,1}  -> float2 load
// C diagonal: element m (m<8) in VGPR m @ lane m; element 8+r in VGPR r @ lane 24+r.

__global__ void dot_wmma_kernel(const int* __restrict__ lsrc, const int* __restrict__ ldst,
                                const float* __restrict__ acc, float* __restrict__ out,
                                int P) {
  int wave = (int)(((long)blockIdx.x * blockDim.x + threadIdx.x) >> 5);
  int lane = threadIdx.x & 31;
  int base = wave * 16;
  if (base >= P) return;  // wave-uniform: surviving waves keep EXEC all-ones for WMMA

  int m = lane & 15;
  int h = lane >> 4;
  int pi = base + m;
  if (pi >= P) pi = P - 1;  // clamp (keeps full-wave EXEC); writes are guarded below
  int rs = lsrc[pi];
  int rd = ldst[pi];
  const float* pS = acc + (long)rs * DD + 2 * h;
  const float* pD = acc + (long)rd * DD + 2 * h;

  v8f c = {};
#pragma unroll
  for (int k0 = 0; k0 < DD; k0 += 4) {
    v2f a = *(const v2f*)(pS + k0);
    v2f b = *(const v2f*)(pD + k0);
    c = __builtin_amdgcn_wmma_f32_16x16x4_f32(false, a, false, b, (short)0, c,
                                              false, false);
  }

  float tmp[8];
#pragma unroll
  for (int j = 0; j < 8; ++j) tmp[j] = c[j];

  if (lane < 8) {
    int p = base + lane;
    if (p < P) out[p] = tmp[lane];
  } else if (lane >= 24) {
    int r = lane - 24;
    int p = base + 8 + r;
    if (p < P) out[p] = tmp[r];
  }
}

// ---------------- launcher ----------------

extern "C" void kernel_launch(void* const* d_in, const int* in_sizes, int n_in,
                              void* d_out, int out_size, void* d_ws, size_t ws_size,
                              hipStream_t stream) {
  const int* edge_index = (const int*)d_in[0];        // [2, E]
  const int* edge_label = (const int*)d_in[1];        // [2, P]
  const float* emb      = (const float*)d_in[2];      // [N, D]

  const int E = in_sizes[0] / 2;
  const int P = in_sizes[1] / 2;
  const long ND = (long)in_sizes[2];                  // N * 128
  const int N = (int)(ND / DD);
  const int L = 3;
  const float w = 1.0f / (float)(L + 1);              // softmax(uniform) -> uniform

  const int* e_src = edge_index;
  const int* e_dst = edge_index + E;
  const int* l_src = edge_label;
  const int* l_dst = edge_label + P;

  // workspace partition (floats): dinv[N] | xA[ND] | xB[ND] | acc[ND]
  float* dinv = (float*)d_ws;
  long npad = ((long)N + 255) & ~255L;
  float* xA  = dinv + npad;
  float* xB  = xA + ND;
  float* acc = xB + ND;

  const int BT = 256;
  long n4 = ND / 4;
  int g_n4   = (int)((n4 + BT - 1) / BT);
  int g_E    = (E + BT - 1) / BT;
  int g_N    = (N + BT - 1) / BT;
  long sthreads = (long)E * 32;
  int g_scat = (int)((sthreads + BT - 1) / BT);

  // degree -> dinv (in place)
  {
    long z4 = (npad) / 4;
    zero4_kernel<<<(int)((z4 + BT - 1) / BT), BT, 0, stream>>>((float4*)dinv, z4);
  }
  deg_kernel<<<g_E, BT, 0, stream>>>(e_dst, dinv, E);
  dinv_kernel<<<g_N, BT, 0, stream>>>(dinv, N);

  // acc = w*emb ; x = emb
  init_kernel<<<g_n4, BT, 0, stream>>>((const float4*)emb, (float4*)xA, (float4*)acc,
                                       n4, w);

  float* xcur = xA;
  float* xnxt = xB;
  for (int layer = 0; layer < L; ++layer) {
    zero4_kernel<<<g_n4, BT, 0, stream>>>((float4*)xnxt, n4);
    scatter_kernel<<<g_scat, BT, 0, stream>>>(e_src, e_dst, dinv, xcur, xnxt, E);
    axpy_kernel<<<g_n4, BT, 0, stream>>>((float4*)acc, (const float4*)xnxt, n4, w);
    float* t = xcur; xcur = xnxt; xnxt = t;
  }

  // link prediction dots (16 pairs per wave, WMMA f32 16x16x4)
  int waves = (P + 15) / 16;
  long dthreads = (long)waves * 32;
  int g_dot = (int)((dthreads + BT - 1) / BT);
  dot_wmma_kernel<<<g_dot, BT, 0, stream>>>(l_src, l_dst, acc, (float*)d_out, P);
}